// tsne_26680336843043
// MI455X (gfx1250) — compile-verified
//
#include <hip/hip_runtime.h>

typedef __attribute__((ext_vector_type(2))) float v2f;
typedef __attribute__((ext_vector_type(8))) float v8f;

#define N_ROWS 8192
#define FEAT   784
#define HID    128
#define KCHUNK 112   // 784 = 7 * 112
#define JCHUNK 4096  // 32KB LDS stage of o2

__device__ __forceinline__ float selu_f(float v) {
    const float kScale      = 1.0507009873554805f;
    const float kScaleAlpha = 1.7580993408473766f; // scale * alpha
    return v > 0.0f ? kScale * v : kScaleAlpha * (__expf(v) - 1.0f);
}

// ---------------------------------------------------------------------------
// Kernel A: o = selu(x @ fc_w.T + fc_b), f32 WMMA 16x16x4.
// One block = 16-row strip of x; 8 waves each own a 16-column slice of the
// 128 output channels. A tile staged in LDS (read once), B (fc_w) is
// L2-resident. Epilogue folds per-channel BN partial sums via f32 atomics.
// ---------------------------------------------------------------------------
__global__ __launch_bounds__(256) void k_gemm1_selu(
    const float* __restrict__ x, const float* __restrict__ w,
    const float* __restrict__ bias, float* __restrict__ o,
    float* __restrict__ s1, float* __restrict__ s2)
{
    __shared__ float lA[16][113];   // 16 rows x 112 k, stride 113 -> conflict-free
    const int t    = threadIdx.x;
    const int lane = t & 31;
    const int wave = t >> 5;
    const int m0   = blockIdx.x * 16;
    const int n0   = wave * 16;
    const int c    = lane & 15;     // N within tile / M row for A-frag
    const int h    = lane >> 4;     // half-wave selects K pair

    v8f acc = {0.f, 0.f, 0.f, 0.f, 0.f, 0.f, 0.f, 0.f};

    const int lrow = t >> 4;        // cooperative A-stage: row 0..15
    const int lcol = t & 15;        // col group, coalesced 64B segments
    const float* wrow = w + (size_t)(n0 + c) * FEAT + 2 * h;

    for (int kb = 0; kb < FEAT; kb += KCHUNK) {
        __syncthreads();
        #pragma unroll
        for (int i = 0; i < KCHUNK / 16; ++i)
            lA[lrow][lcol + 16 * i] = x[(size_t)(m0 + lrow) * FEAT + kb + lcol + 16 * i];
        __syncthreads();

        #pragma unroll
        for (int kk = 0; kk < KCHUNK; kk += 4) {
            v2f a, b;
            a.x = lA[c][kk + 2 * h];
            a.y = lA[c][kk + 2 * h + 1];
            const float* bp = wrow + kb + kk;
            b.x = bp[0];
            b.y = bp[1];
            acc = __builtin_amdgcn_wmma_f32_16x16x4_f32(
                false, a, false, b, (short)0, acc, false, false);
        }
    }

    const float bv = bias[n0 + c];
    float p1 = 0.f, p2 = 0.f;
    #pragma unroll
    for (int r = 0; r < 8; ++r) {
        float v = selu_f(acc[r] + bv);
        o[(size_t)(m0 + r + 8 * h) * HID + (n0 + c)] = v;
        p1 += v;
        p2 += v * v;
    }
    atomicAdd(&s1[n0 + c], p1);
    atomicAdd(&s2[n0 + c], p2);
}

// ---------------------------------------------------------------------------
// Kernel B: finalize BN per-channel scale/shift:  bn(v) = v*a + b
// ---------------------------------------------------------------------------
__global__ void k_bn_finalize(
    const float* __restrict__ s1, const float* __restrict__ s2,
    const float* __restrict__ gamma, const float* __restrict__ beta,
    float* __restrict__ bnA, float* __restrict__ bnB)
{
    int cid = threadIdx.x;
    float mean = s1[cid] * (1.0f / N_ROWS);
    float var  = fmaxf(s2[cid] * (1.0f / N_ROWS) - mean * mean, 0.0f);
    float a    = gamma[cid] * rsqrtf(var + 1e-5f);
    bnA[cid] = a;
    bnB[cid] = beta[cid] - mean * a;
}

// ---------------------------------------------------------------------------
// Kernel C: o2 = selu(bn(o) @ fc2_w.T + fc2_b)   (d=2 -> plain VALU dot)
// ---------------------------------------------------------------------------
__global__ __launch_bounds__(256) void k_fc2_selu(
    const float* __restrict__ o, const float* __restrict__ bnA,
    const float* __restrict__ bnB, const float* __restrict__ w2,
    const float* __restrict__ b2, float2* __restrict__ o2,
    float2* __restrict__ o_out)
{
    __shared__ float sA[HID], sB[HID], sw0[HID], sw1[HID];
    int t = threadIdx.x;
    if (t < HID) {
        sA[t]  = bnA[t];
        sB[t]  = bnB[t];
        sw0[t] = w2[t];
        sw1[t] = w2[HID + t];
    }
    __syncthreads();

    int m = blockIdx.x * 256 + t;
    float a0 = b2[0], a1 = b2[1];
    const float* orow = o + (size_t)m * HID;
    #pragma unroll 4
    for (int k = 0; k < HID; ++k) {
        float v = fmaf(orow[k], sA[k], sB[k]);
        a0 = fmaf(v, sw0[k], a0);
        a1 = fmaf(v, sw1[k], a1);
    }
    float2 r;
    r.x = selu_f(a0);
    r.y = selu_f(a1);
    o2[m]    = r;
    o_out[m] = r;   // second reference output
}

// ---------------------------------------------------------------------------
// Kernel D: global sum of 1/(1+dis_ij) over all pairs (includes diag = +n).
// Block = 16 i-rows x all j; o2 staged in 32KB LDS chunks.
// ---------------------------------------------------------------------------
__global__ __launch_bounds__(256) void k_qsum(
    const float2* __restrict__ o2, double* __restrict__ denom)
{
    __shared__ float2 s[JCHUNK];
    __shared__ float red[256];
    int t  = threadIdx.x;
    int i0 = blockIdx.x * 16;
    float lsum = 0.f;

    for (int jb = 0; jb < N_ROWS; jb += JCHUNK) {
        __syncthreads();
        #pragma unroll
        for (int k = 0; k < JCHUNK / 256; ++k)
            s[t + 256 * k] = o2[jb + t + 256 * k];
        __syncthreads();
        for (int ii = 0; ii < 16; ++ii) {
            float2 oi = o2[i0 + ii];
            #pragma unroll 4
            for (int j = t; j < JCHUNK; j += 256) {
                float dx = oi.x - s[j].x;
                float dy = oi.y - s[j].y;
                float d  = fmaf(dx, dx, fmaf(dy, dy, 1.0f));
                lsum += __builtin_amdgcn_rcpf(d);
            }
        }
    }

    red[t] = lsum;
    __syncthreads();
    for (int off = 128; off > 0; off >>= 1) {
        if (t < off) red[t] += red[t + off];
        __syncthreads();
    }
    if (t == 0) atomicAdd(denom, (double)red[0]);
}

// ---------------------------------------------------------------------------
// Kernel E: qij = (1/(1+dis_ij)) * (1/(sum - n)); 256 MB coalesced NT stream.
// ---------------------------------------------------------------------------
__global__ __launch_bounds__(256) void k_qwrite(
    const float2* __restrict__ o2, const double* __restrict__ denom,
    float* __restrict__ qij)
{
    __shared__ float2 s[JCHUNK];
    int t  = threadIdx.x;
    int i0 = blockIdx.x * 16;
    float invd = (float)(1.0 / (*denom - (double)N_ROWS));

    for (int jb = 0; jb < N_ROWS; jb += JCHUNK) {
        __syncthreads();
        #pragma unroll
        for (int k = 0; k < JCHUNK / 256; ++k)
            s[t + 256 * k] = o2[jb + t + 256 * k];
        __syncthreads();
        for (int ii = 0; ii < 16; ++ii) {
            float2 oi = o2[i0 + ii];
            float* qrow = qij + (size_t)(i0 + ii) * N_ROWS + jb;
            #pragma unroll 4
            for (int j = t; j < JCHUNK; j += 256) {
                float dx = oi.x - s[j].x;
                float dy = oi.y - s[j].y;
                float d  = fmaf(dx, dx, fmaf(dy, dy, 1.0f));
                __builtin_nontemporal_store(__builtin_amdgcn_rcpf(d) * invd, qrow + j);
            }
        }
    }
}

// ---------------------------------------------------------------------------
extern "C" void kernel_launch(void* const* d_in, const int* in_sizes, int n_in,
                              void* d_out, int out_size, void* d_ws, size_t ws_size,
                              hipStream_t stream)
{
    const float* x     = (const float*)d_in[0];
    const float* fc_w  = (const float*)d_in[1];
    const float* fc_b  = (const float*)d_in[2];
    const float* gamma = (const float*)d_in[3];
    const float* beta  = (const float*)d_in[4];
    const float* fc2_w = (const float*)d_in[5];
    const float* fc2_b = (const float*)d_in[6];

    float*  qij   = (float*)d_out;
    float2* o_out = (float2*)(qij + (size_t)N_ROWS * N_ROWS);

    // workspace layout (all offsets keep >= 8B alignment)
    float*  ws  = (float*)d_ws;
    float*  o   = ws;                        // 8192*128 f32 (4 MB)
    float*  s1  = o + (size_t)N_ROWS * HID;  // 128
    float*  s2  = s1 + HID;                  // 128
    float*  bnA = s2 + HID;                  // 128
    float*  bnB = bnA + HID;                 // 128
    float2* o2  = (float2*)(bnB + HID);      // 8192 float2 (64 KB)
    double* den = (double*)(o2 + N_ROWS);    // 1 double

    hipMemsetAsync(s1, 0, 2 * HID * sizeof(float), stream);
    hipMemsetAsync(den, 0, sizeof(double), stream);

    k_gemm1_selu<<<N_ROWS / 16, 256, 0, stream>>>(x, fc_w, fc_b, o, s1, s2);
    k_bn_finalize<<<1, HID, 0, stream>>>(s1, s2, gamma, beta, bnA, bnB);
    k_fc2_selu<<<N_ROWS / 256, 256, 0, stream>>>(o, bnA, bnB, fc2_w, fc2_b, o2, o_out);
    k_qsum<<<N_ROWS / 16, 256, 0, stream>>>(o2, den);
    k_qwrite<<<N_ROWS / 16, 256, 0, stream>>>(o2, den, qij);
}